// SceneGNN_4088808866429
// MI455X (gfx1250) — compile-verified
//
#include <hip/hip_runtime.h>

#define N_NODES 100000
#define N_EDGES 1600000
#define DIM 128
#define N_GRAPHS 512

typedef float v2f __attribute__((ext_vector_type(2)));
typedef float v8f __attribute__((ext_vector_type(8)));
typedef unsigned int u32x4 __attribute__((ext_vector_type(4)));
typedef int i32x4 __attribute__((ext_vector_type(4)));
typedef int i32x8 __attribute__((ext_vector_type(8)));

// ---------------------------------------------------------------------------
// GEMM: C[M x 128] = A[M x 128] * W[128 x 128] using V_WMMA_F32_16X16X4_F32.
// One block = 256 threads = 8 waves. W staged in 64KB LDS via the Tensor Data
// Mover (one DMA descriptor issued by wave 0, tracked by TENSORcnt), instead
// of 4096 vector loads+stores. Each wave owns a 16-row x 128-col output
// stripe (8 v8f accumulators), K-loop of 32 WMMA steps.
// A-frag (ISA 7.12.2, 32-bit A 16x4): lane&15 = M; (lane>>4)*2 = K-half;
// VGPR0 = K+0, VGPR1 = K+1 -> one aligned float2 load.
// B-frag (4x16): lane&15 = N; VGPR0 = W[k+half][n], VGPR1 = W[k+half+1][n].
// C/D: VGPR r -> M = r + 8*(lane>>4), N = lane&15.
// ---------------------------------------------------------------------------
__global__ __launch_bounds__(256) void gemm_wmma_f32(
    const float* __restrict__ A, const float* __restrict__ W,
    float* __restrict__ C, int M) {
  __shared__ float sW[DIM * DIM];  // 64 KB

  // --- TDM stage: wave 0 issues one tensor_load_to_lds for all of W -------
  if (threadIdx.x < 32) {
    unsigned lds_off = (unsigned)(size_t)(void*)sW;  // LDS aperture: addr[31:0] = LDS offset
    unsigned long long wa = (unsigned long long)W;
    // D# group 0: count=1 | lds_addr | global_addr[56:0] | type=2
    u32x4 g0;
    g0[0] = 1u;                                   // count=1, user desc
    g0[1] = lds_off;                              // lds_addr
    g0[2] = (unsigned)wa;                         // global_addr[31:0]
    g0[3] = (unsigned)((wa >> 32) & 0x01FFFFFFu) | (2u << 30);  // addr[56:32] | type=2
    // D# group 1: 2D tensor 128x128, tile 128x128, data_size=4B, stride=128
    i32x8 g1;
    g1[0] = 2 << 16;        // workgroup_mask=0, data_size=2 (4 bytes)
    g1[1] = DIM << 16;      // tensor_dim0[15:0] at bits[63:48]
    g1[2] = DIM << 16;      // tensor_dim0[31:16]=0 | tensor_dim1[15:0]
    g1[3] = DIM << 16;      // tensor_dim1[31:16]=0 | tile_dim0=128
    g1[4] = DIM;            // tile_dim1=128, tile_dim2=0
    g1[5] = DIM;            // tensor_dim0_stride[31:0] = 128 elements
    g1[6] = 0;              // stride hi | tensor_dim1_stride lo
    g1[7] = 0;
    i32x4 gz4 = (i32x4){0, 0, 0, 0};              // groups 2/3 unused (<=2D tensor)
    i32x8 gz8 = (i32x8){0, 0, 0, 0, 0, 0, 0, 0};  // extra group (clang-23 6-arg form)
    __builtin_amdgcn_tensor_load_to_lds(g0, g1, gz4, gz4, gz8, 0);
    __builtin_amdgcn_s_wait_tensorcnt(0);
  }
  __syncthreads();

  const int tid = threadIdx.x;
  const int wave = tid >> 5;
  const int lane = tid & 31;
  const int tile = blockIdx.x * 8 + wave;
  if (tile * 16 >= M) return;  // whole wave exits uniformly (EXEC stays full)

  const int row0  = tile * 16;
  const int mrow  = row0 + (lane & 15);
  const int khalf = (lane >> 4) * 2;  // 0 or 2
  const int ncol  = lane & 15;

  v8f acc[8];
  #pragma unroll
  for (int n = 0; n < 8; ++n) acc[n] = (v8f){0.f,0.f,0.f,0.f,0.f,0.f,0.f,0.f};

  #pragma unroll 4
  for (int kk = 0; kk < DIM; kk += 4) {
    const v2f a = *(const v2f*)(A + (size_t)mrow * DIM + kk + khalf);
    #pragma unroll
    for (int n = 0; n < 8; ++n) {
      const float* bp = sW + (size_t)(kk + khalf) * DIM + n * 16 + ncol;
      v2f b;
      b.x = bp[0];
      b.y = bp[DIM];
      acc[n] = __builtin_amdgcn_wmma_f32_16x16x4_f32(
          /*neg_a=*/false, a, /*neg_b=*/false, b,
          /*c_mod=*/(short)0, acc[n], /*reuse_a=*/false, /*reuse_b=*/false);
    }
  }

  const int rowBase = row0 + (lane >> 4) * 8;
  #pragma unroll
  for (int n = 0; n < 8; ++n) {
    float* cp = C + (size_t)rowBase * DIM + n * 16 + ncol;
    #pragma unroll
    for (int r = 0; r < 8; ++r) cp[(size_t)r * DIM] = acc[n][r];
  }
}

// ---------------------------------------------------------------------------
// Elementwise / scatter helpers
// ---------------------------------------------------------------------------
__global__ void zero_f32(float* __restrict__ p, int n) {
  int i = blockIdx.x * blockDim.x + threadIdx.x;
  if (i < n) p[i] = 0.0f;
}

__global__ void fill_ones(float* __restrict__ p, int n) {
  int i = blockIdx.x * blockDim.x + threadIdx.x;
  if (i < n) p[i] = 1.0f;  // deg starts at 1 (self-loop)
}

__global__ void deg_count(const int* __restrict__ dst, float* __restrict__ deg, int nE) {
  int e = blockIdx.x * blockDim.x + threadIdx.x;
  if (e < nE) atomicAdd(&deg[dst[e]], 1.0f);
}

__global__ void deg_to_dinv(float* __restrict__ deg, int n) {
  int i = blockIdx.x * blockDim.x + threadIdx.x;
  if (i < n) deg[i] = rsqrtf(deg[i]);
}

// 32 lanes per edge; each lane handles 4 consecutive features (float4 gather,
// 4x global_atomic_add_f32 scatter). Row = 512B -> fully coalesced per wave.
__global__ __launch_bounds__(256) void scatter_edges(
    const float* __restrict__ h, const int* __restrict__ src,
    const int* __restrict__ dst, const float* __restrict__ dinv,
    float* __restrict__ agg, int nE) {
  int t = blockIdx.x * blockDim.x + threadIdx.x;
  int e = t >> 5;
  if (e >= nE) return;
  int lane = t & 31;
  int s = src[e], d = dst[e];
  float norm = dinv[s] * dinv[d];
  float4 hv = ((const float4*)(h + (size_t)s * DIM))[lane];
  float* out = agg + (size_t)d * DIM + lane * 4;
  atomicAdd(out + 0, hv.x * norm);
  atomicAdd(out + 1, hv.y * norm);
  atomicAdd(out + 2, hv.z * norm);
  atomicAdd(out + 3, hv.w * norm);
}

// act = relu(agg + h*dinv^2 + bias); re-zeros agg in the same pass so the
// next layer's scatter can reuse it without an extra sweep.
__global__ void finalize_relu(const float* __restrict__ agg, float* __restrict__ agg_clear,
                              const float* __restrict__ h, const float* __restrict__ dinv,
                              const float* __restrict__ bias, float* __restrict__ act, int n) {
  int t = blockIdx.x * blockDim.x + threadIdx.x;
  if (t >= n * DIM) return;
  int i = t >> 7, f = t & (DIM - 1);
  float di = dinv[i];
  float v = agg[t] + h[t] * di * di + bias[f];
  agg_clear[t] = 0.0f;
  act[t] = fmaxf(v, 0.0f);
}

// Layer-2 finalize fused with segment-sum pooling.
__global__ void finalize_pool(const float* __restrict__ agg, const float* __restrict__ h,
                              const float* __restrict__ dinv, const float* __restrict__ bias,
                              const int* __restrict__ batch, float* __restrict__ pool,
                              float* __restrict__ counts, int n) {
  int t = blockIdx.x * blockDim.x + threadIdx.x;
  if (t >= n * DIM) return;
  int i = t >> 7, f = t & (DIM - 1);
  float di = dinv[i];
  float v = fmaxf(agg[t] + h[t] * di * di + bias[f], 0.0f);
  int g = batch[i];
  atomicAdd(&pool[(size_t)g * DIM + f], v);
  if (f == 0) atomicAdd(&counts[g], 1.0f);
}

__global__ void pool_div(const float* __restrict__ pool, const float* __restrict__ counts,
                         float* __restrict__ out) {
  int t = blockIdx.x * blockDim.x + threadIdx.x;
  if (t >= N_GRAPHS * DIM) return;
  int g = t >> 7;
  float c = counts[g];
  out[t] = (c > 0.0f) ? pool[t] / fmaxf(c, 1.0f) : 0.0f;
}

// ---------------------------------------------------------------------------
extern "C" void kernel_launch(void* const* d_in, const int* in_sizes, int n_in,
                              void* d_out, int out_size, void* d_ws, size_t ws_size,
                              hipStream_t stream) {
  const float* x   = (const float*)d_in[0];
  const float* W1  = (const float*)d_in[1];
  const float* b1  = (const float*)d_in[2];
  const float* W2  = (const float*)d_in[3];
  const float* b2  = (const float*)d_in[4];
  const int* edge  = (const int*)d_in[5];
  const int* batch = (const int*)d_in[6];
  const int* src = edge;
  const int* dst = edge + N_EDGES;
  float* out = (float*)d_out;

  const size_t NF = (size_t)N_NODES * DIM;  // 12.8M floats
  char* ws = (char*)d_ws;
  float* h    = (float*)ws; ws += NF * sizeof(float);
  float* agg  = (float*)ws; ws += NF * sizeof(float);
  float* act  = (float*)ws; ws += NF * sizeof(float);
  float* dinv = (float*)ws; ws += ((size_t)N_NODES + 32) * sizeof(float);
  float* pool = (float*)ws; ws += (size_t)N_GRAPHS * DIM * sizeof(float);
  float* cnts = (float*)ws;

  const int B = 256;
  const int nodeBlocks = (N_NODES + B - 1) / B;           // 391
  const int featBlocks = (int)((NF + B - 1) / B);         // 50000
  const int edgeBlocks = (N_EDGES + B - 1) / B;           // 6250
  const int scatBlocks = (int)(((size_t)N_EDGES * 32 + B - 1) / B);  // 200000
  const int gemmBlocks = (N_NODES / 16 + 7) / 8;          // 782 (6250 tiles / 8 waves)

  // init: deg=1 (self-loop), agg=0, pool/counts=0
  fill_ones<<<nodeBlocks, B, 0, stream>>>(dinv, N_NODES);
  zero_f32<<<featBlocks, B, 0, stream>>>(agg, (int)NF);
  zero_f32<<<(N_GRAPHS * DIM + B - 1) / B, B, 0, stream>>>(pool, N_GRAPHS * DIM);
  zero_f32<<<(N_GRAPHS + B - 1) / B, B, 0, stream>>>(cnts, N_GRAPHS);

  // degree -> dinv
  deg_count<<<edgeBlocks, B, 0, stream>>>(dst, dinv, N_EDGES);
  deg_to_dinv<<<nodeBlocks, B, 0, stream>>>(dinv, N_NODES);

  // layer 1
  gemm_wmma_f32<<<gemmBlocks, B, 0, stream>>>(x, W1, h, N_NODES);
  scatter_edges<<<scatBlocks, B, 0, stream>>>(h, src, dst, dinv, agg, N_EDGES);
  finalize_relu<<<featBlocks, B, 0, stream>>>(agg, agg, h, dinv, b1, act, N_NODES);

  // layer 2 (agg was re-zeroed by finalize_relu)
  gemm_wmma_f32<<<gemmBlocks, B, 0, stream>>>(act, W2, h, N_NODES);
  scatter_edges<<<scatBlocks, B, 0, stream>>>(h, src, dst, dinv, agg, N_EDGES);
  finalize_pool<<<featBlocks, B, 0, stream>>>(agg, h, dinv, b2, batch, pool, cnts, N_NODES);

  // mean pool
  pool_div<<<(N_GRAPHS * DIM + B - 1) / B, B, 0, stream>>>(pool, cnts, out);
}